// PRMPConvFull_1099511628128
// MI455X (gfx1250) — compile-verified
//
#include <hip/hip_runtime.h>

typedef __attribute__((ext_vector_type(2))) float v2f;
typedef __attribute__((ext_vector_type(8))) float v8f;

#define D_DIM 64
#define WPITCH 66   // LDS row pitch in floats: avoids 16-way bank conflicts, keeps 8B align

static __device__ __forceinline__ v8f wmma_f32(v2f a, v2f b, v8f c) {
  // D = A(16x4 f32) * B(4x16 f32) + C(16x16 f32), exact fp32 math
  return __builtin_amdgcn_wmma_f32_16x16x4_f32(false, a, false, b, (short)0, c,
                                               false, false);
}

// ---------------------------------------------------------------------------
// Kernel 0: zero sums[] and count[] in workspace
// ---------------------------------------------------------------------------
__global__ void zero_kernel(float* __restrict__ p, int n) {
  int i = blockIdx.x * blockDim.x + threadIdx.x;
  if (i < n) p[i] = 0.0f;
}

// ---------------------------------------------------------------------------
// Kernel 1: pred_src = relu(x_dst @ W1^T + b1) @ W2^T + b2
// 128 threads = 4 waves; each wave owns one 16-row tile. Both GEMMs fused,
// h staged through LDS between them (C-layout -> A-layout transpose).
// ---------------------------------------------------------------------------
__global__ __launch_bounds__(128) void pred_mlp_kernel(
    const float* __restrict__ xdst,
    const float* __restrict__ W1, const float* __restrict__ b1,
    const float* __restrict__ W2, const float* __restrict__ b2,
    float* __restrict__ pred, int nD, int nTiles) {
  __shared__ float sW1[D_DIM * WPITCH];
  __shared__ float sW2[D_DIM * WPITCH];
  __shared__ float sX[4][16 * WPITCH];

  const int tid  = threadIdx.x;
  const int w    = tid >> 5;
  const int lane = tid & 31;
  const int half = lane >> 4;
  const int l    = lane & 15;

  // Cooperative weight load: W[n][k] -> s[n*WPITCH+k]  (coalesced global reads)
  for (int idx = tid; idx < D_DIM * D_DIM; idx += 128) {
    int n = idx >> 6, k = idx & 63;
    sW1[n * WPITCH + k] = W1[idx];
    sW2[n * WPITCH + k] = W2[idx];
  }

  const int tile = blockIdx.x * 4 + w;
  const int tb   = tile * 16;
  if (tile < nTiles) {
    for (int idx = lane; idx < 16 * D_DIM; idx += 32) {
      int m = idx >> 6, k = idx & 63;
      int row = tb + m; if (row >= nD) row = nD - 1;
      sX[w][m * WPITCH + k] = xdst[(size_t)row * D_DIM + k];
    }
  }
  __syncthreads();
  if (tile >= nTiles) return;                    // wave-uniform exit

  float* sXw = sX[w];

  // GEMM 1: h = relu(X @ W1^T + b1)   A[m][k]=X, B[k][n]=W1[n][k]
  v8f acc[4];
  #pragma unroll
  for (int nb = 0; nb < 4; ++nb) {
    float bv = b1[nb * 16 + l];
    #pragma unroll
    for (int r = 0; r < 8; ++r) acc[nb][r] = bv;
  }
  #pragma unroll
  for (int k = 0; k < D_DIM; k += 4) {
    const int kk = k + 2 * half;
    v2f a = *(const v2f*)&sXw[l * WPITCH + kk];
    #pragma unroll
    for (int nb = 0; nb < 4; ++nb) {
      v2f b = *(const v2f*)&sW1[(nb * 16 + l) * WPITCH + kk];
      acc[nb] = wmma_f32(a, b, acc[nb]);
    }
  }
  // ReLU + restage h (C-layout: N=lane, M over vgprs) into LDS as h[m][n]
  #pragma unroll
  for (int nb = 0; nb < 4; ++nb) {
    #pragma unroll
    for (int r = 0; r < 8; ++r) {
      float v = acc[nb][r];
      v = v > 0.0f ? v : 0.0f;
      sXw[(r + 8 * half) * WPITCH + nb * 16 + l] = v;
    }
  }

  // GEMM 2: pred = h @ W2^T + b2
  #pragma unroll
  for (int nb = 0; nb < 4; ++nb) {
    float bv = b2[nb * 16 + l];
    #pragma unroll
    for (int r = 0; r < 8; ++r) acc[nb][r] = bv;
  }
  #pragma unroll
  for (int k = 0; k < D_DIM; k += 4) {
    const int kk = k + 2 * half;
    v2f a = *(const v2f*)&sXw[l * WPITCH + kk];
    #pragma unroll
    for (int nb = 0; nb < 4; ++nb) {
      v2f b = *(const v2f*)&sW2[(nb * 16 + l) * WPITCH + kk];
      acc[nb] = wmma_f32(a, b, acc[nb]);
    }
  }
  #pragma unroll
  for (int nb = 0; nb < 4; ++nb) {
    #pragma unroll
    for (int r = 0; r < 8; ++r) {
      int row = tb + r + 8 * half;
      if (row < nD) pred[(size_t)row * D_DIM + nb * 16 + l] = acc[nb][r];
    }
  }
}

// ---------------------------------------------------------------------------
// Kernel 2: per-edge residual scatter-add. One wave per edge; each lane owns
// two features (float2). sums[] is 12.8MB -> resident in L2, atomics are cheap.
// ---------------------------------------------------------------------------
__global__ __launch_bounds__(256) void edge_scatter_kernel(
    const float* __restrict__ x_src, const float* __restrict__ pred,
    const int* __restrict__ esrc, const int* __restrict__ edst,
    float* __restrict__ sums, float* __restrict__ cnt, int nE) {
  const int wid  = (blockIdx.x * blockDim.x + threadIdx.x) >> 5;
  const int lane = threadIdx.x & 31;
  if (wid >= nE) return;
  const int s = esrc[wid];
  const int d = edst[wid];
  const float2 a = ((const float2*)(x_src + (size_t)s * D_DIM))[lane];
  const float2 p = ((const float2*)(pred  + (size_t)d * D_DIM))[lane];
  float* dst = sums + (size_t)d * D_DIM + 2 * lane;
  atomicAdd(dst,     a.x - p.x);
  atomicAdd(dst + 1, a.y - p.y);
  if (lane == 0) atomicAdd(&cnt[d], 1.0f);
}

// ---------------------------------------------------------------------------
// Kernel 3: out = (sums/max(cnt,1)) @ Wn^T + bn + x_dst @ Ws^T + bs
// Two GEMMs accumulated into the same WMMA C fragment; biases folded into init.
// ---------------------------------------------------------------------------
__global__ __launch_bounds__(128) void out_kernel(
    const float* __restrict__ xdst,
    const float* __restrict__ sums, const float* __restrict__ cnt,
    const float* __restrict__ Wn, const float* __restrict__ bn,
    const float* __restrict__ Ws, const float* __restrict__ bs,
    float* __restrict__ out, int nD, int nTiles) {
  __shared__ float sWn[D_DIM * WPITCH];
  __shared__ float sWs[D_DIM * WPITCH];
  __shared__ float sA[4][16 * WPITCH];
  __shared__ float sX[4][16 * WPITCH];

  const int tid  = threadIdx.x;
  const int w    = tid >> 5;
  const int lane = tid & 31;
  const int half = lane >> 4;
  const int l    = lane & 15;

  for (int idx = tid; idx < D_DIM * D_DIM; idx += 128) {
    int n = idx >> 6, k = idx & 63;
    sWn[n * WPITCH + k] = Wn[idx];
    sWs[n * WPITCH + k] = Ws[idx];
  }

  const int tile = blockIdx.x * 4 + w;
  const int tb   = tile * 16;
  if (tile < nTiles) {
    for (int idx = lane; idx < 16 * D_DIM; idx += 32) {
      int m = idx >> 6, k = idx & 63;
      int row = tb + m; if (row >= nD) row = nD - 1;
      float inv = 1.0f / fmaxf(cnt[row], 1.0f);
      sA[w][m * WPITCH + k] = sums[(size_t)row * D_DIM + k] * inv;
      sX[w][m * WPITCH + k] = xdst[(size_t)row * D_DIM + k];
    }
  }
  __syncthreads();
  if (tile >= nTiles) return;

  const float* sAw = sA[w];
  const float* sXw = sX[w];

  v8f acc[4];
  #pragma unroll
  for (int nb = 0; nb < 4; ++nb) {
    float bv = bn[nb * 16 + l] + bs[nb * 16 + l];
    #pragma unroll
    for (int r = 0; r < 8; ++r) acc[nb][r] = bv;
  }
  #pragma unroll
  for (int k = 0; k < D_DIM; k += 4) {
    const int kk = k + 2 * half;
    v2f aAgg = *(const v2f*)&sAw[l * WPITCH + kk];
    v2f aX   = *(const v2f*)&sXw[l * WPITCH + kk];
    #pragma unroll
    for (int nb = 0; nb < 4; ++nb) {
      v2f bN = *(const v2f*)&sWn[(nb * 16 + l) * WPITCH + kk];
      v2f bS = *(const v2f*)&sWs[(nb * 16 + l) * WPITCH + kk];
      acc[nb] = wmma_f32(aAgg, bN, acc[nb]);
      acc[nb] = wmma_f32(aX,   bS, acc[nb]);
    }
  }
  #pragma unroll
  for (int nb = 0; nb < 4; ++nb) {
    #pragma unroll
    for (int r = 0; r < 8; ++r) {
      int row = tb + r + 8 * half;
      if (row < nD) out[(size_t)row * D_DIM + nb * 16 + l] = acc[nb][r];
    }
  }
}

// ---------------------------------------------------------------------------
extern "C" void kernel_launch(void* const* d_in, const int* in_sizes, int n_in,
                              void* d_out, int out_size, void* d_ws, size_t ws_size,
                              hipStream_t stream) {
  const float* x_src = (const float*)d_in[0];
  const float* x_dst = (const float*)d_in[1];
  const int*   esrc  = (const int*)d_in[2];
  const int*   edst  = (const int*)d_in[3];
  // d_in[4] = num_dst (device scalar) -- derived from in_sizes instead
  const float* W1 = (const float*)d_in[5];
  const float* b1 = (const float*)d_in[6];
  const float* W2 = (const float*)d_in[7];
  const float* b2 = (const float*)d_in[8];
  const float* Wn = (const float*)d_in[9];
  const float* bn = (const float*)d_in[10];
  const float* Ws = (const float*)d_in[11];
  const float* bs = (const float*)d_in[12];

  const int nD = in_sizes[1] / D_DIM;
  const int nE = in_sizes[2];

  float* sums = (float*)d_ws;                  // [nD * 64]
  float* cnt  = sums + (size_t)nD * D_DIM;     // [nD]
  float* pred = cnt + nD;                      // [nD * 64]

  const int nzero = nD * (D_DIM + 1);          // sums + cnt are contiguous
  zero_kernel<<<(nzero + 255) / 256, 256, 0, stream>>>(sums, nzero);

  const int nTiles = (nD + 15) / 16;
  const int gemmBlocks = (nTiles + 3) / 4;

  pred_mlp_kernel<<<gemmBlocks, 128, 0, stream>>>(x_dst, W1, b1, W2, b2, pred,
                                                  nD, nTiles);
  edge_scatter_kernel<<<(nE + 7) / 8, 256, 0, stream>>>(x_src, pred, esrc, edst,
                                                        sums, cnt, nE);
  out_kernel<<<gemmBlocks, 128, 0, stream>>>(x_dst, sums, cnt, Wn, bn, Ws, bs,
                                             (float*)d_out, nD, nTiles);
}